// CRFExtensionModule_29540785061994
// MI455X (gfx1250) — compile-verified
//
#include <hip/hip_runtime.h>
#include <hip/hip_bf16.h>

// ---------------------------------------------------------------------------
// CRF mean-field on MI455X (gfx1250, wave32).
// Pipeline (all intermediates L2-resident, ~100 MB < 192 MB L2):
//   conv3x3 -> unary (ws)
//   5x { vblur_softmax: cur -> tmp   (softmax fused into WMMA B-fragments)
//        hblur_add:     tmp + unary -> cur(d_out) }
// Blur-as-matmul: D(16x16) = Gband(16x28) x Window(28x16) using
// V_WMMA_F32_16X16X4_F32 (7 K-steps).
// ---------------------------------------------------------------------------

typedef __attribute__((ext_vector_type(2))) float v2f;
typedef __attribute__((ext_vector_type(8))) float v8f;

#define Hh 512
#define Ww 512
#define HW (512 * 512)
#define CHW (2 * 512 * 512)

__device__ __forceinline__ v8f wmma4(v2f a, v2f b, v8f c) {
  // (neg_a, A, neg_b, B, c_mod, C, reuse_a, reuse_b)
  return __builtin_amdgcn_wmma_f32_16x16x4_f32(false, a, false, b, (short)0, c,
                                               false, false);
}

// Normalized 11-tap Gaussian, theta=1: g[d] = exp(-(d-5)^2/2) / 2.50662824
__device__ __forceinline__ float gband(int m, int k) {
  int d = k - m;                 // band: valid taps at 0..10
  if (d < 0 || d > 10) return 0.0f;
  float t = (float)(d - 5);
  return 0.3989423f * __expf(-0.5f * t * t);
}

// ---------------------------------------------------------------------------
// Kernel 1: 3x3 conv, SAME zero padding. One thread per pixel, both C_out.
// ---------------------------------------------------------------------------
__global__ void __launch_bounds__(256)
conv3x3_kernel(const float* __restrict__ x, const float* __restrict__ w,
               const float* __restrict__ bias, float* __restrict__ unary) {
  int gid = blockIdx.x * 256 + threadIdx.x;      // 16*512*512 threads
  int px = gid & 511;
  int py = (gid >> 9) & 511;
  int b  = gid >> 18;
  const float* xb = x + (size_t)b * 3 * HW;

  float acc0 = bias[0], acc1 = bias[1];
#pragma unroll
  for (int ci = 0; ci < 3; ++ci) {
#pragma unroll
    for (int ky = 0; ky < 3; ++ky) {
      int yy = py + ky - 1;
      if ((unsigned)yy < 512u) {
#pragma unroll
        for (int kx = 0; kx < 3; ++kx) {
          int xx = px + kx - 1;
          if ((unsigned)xx < 512u) {
            float v = xb[ci * HW + yy * Ww + xx];
            acc0 = fmaf(v, w[(0 * 3 + ci) * 9 + ky * 3 + kx], acc0);
            acc1 = fmaf(v, w[(1 * 3 + ci) * 9 + ky * 3 + kx], acc1);
          }
        }
      }
    }
  }
  size_t base = (size_t)b * CHW + (size_t)py * Ww + px;
  unary[base]      = acc0;
  unary[base + HW] = acc1;
}

// ---------------------------------------------------------------------------
// Kernel 2: fused softmax + vertical blur.  One wave per 16x16 tile.
//   D = Gband(16x28) x Q(28x16), Q built per-lane from softmax(cur).
// A frag (16x4 f32): lane m=l&15; vgpr0={K0,K2-half}, vgpr1={K1,K3-half}.
// B frag (4x16 f32): lanes 0-15 hold K={k0,k0+1}, lanes 16-31 K={k0+2,k0+3}.
// ---------------------------------------------------------------------------
__global__ void __launch_bounds__(256)
crf_vblur_softmax(const float* __restrict__ src, float* __restrict__ tmp) {
  const int lane  = threadIdx.x & 31;
  const int wid   = blockIdx.x * 8 + (threadIdx.x >> 5);
  const int tc    = wid & 31;
  const int tr    = (wid >> 5) & 31;
  const int b     = wid >> 10;
  const int m     = lane & 15;
  const int khalf = (lane < 16) ? 0 : 2;
  const int col   = tc * 16 + m;

  const float* s0 = src + (size_t)b * CHW;
  const float* s1 = s0 + HW;

  v2f A[7];
#pragma unroll
  for (int s = 0; s < 7; ++s) {
    int k0 = 4 * s + khalf;
    A[s].x = gband(m, k0);
    A[s].y = gband(m, k0 + 1);
  }

  v8f acc0 = {0.f, 0.f, 0.f, 0.f, 0.f, 0.f, 0.f, 0.f};
  v8f acc1 = acc0;
  const int rbase = tr * 16 - 5;   // window rows r0-5 .. r0+20, K padded to 28

#pragma unroll
  for (int s = 0; s < 7; ++s) {
    v2f B0, B1;
#pragma unroll
    for (int p = 0; p < 2; ++p) {
      int k   = 4 * s + khalf + p;
      int row = rbase + k;
      float q0 = 0.f, q1 = 0.f;
      if (k < 26 && (unsigned)row < 512u) {   // zero pad + K-pad (A col is 0)
        float c0 = s0[row * Ww + col];
        float c1 = s1[row * Ww + col];
        float mx = fmaxf(c0, c1);
        float e0 = __expf(c0 - mx);
        float e1 = __expf(c1 - mx);
        // sum in [1,2]; 1-ulp v_rcp_f32 instead of IEEE divide sequence
        float inv = __builtin_amdgcn_rcpf(e0 + e1);
        q0 = e0 * inv;
        q1 = e1 * inv;
      }
      B0[p] = q0;
      B1[p] = q1;
    }
    acc0 = wmma4(A[s], B0, acc0);
    acc1 = wmma4(A[s], B1, acc1);
  }

  // D layout: vgpr v -> row m = v + (lane<16 ? 0 : 8), col = lane&15
  float* t0 = tmp + (size_t)b * CHW;
  float* t1 = t0 + HW;
  const int rout = tr * 16 + ((lane < 16) ? 0 : 8);
#pragma unroll
  for (int v = 0; v < 8; ++v) {
    t0[(rout + v) * Ww + col] = acc0[v];
    t1[(rout + v) * Ww + col] = acc1[v];
  }
}

// ---------------------------------------------------------------------------
// Kernel 3: horizontal blur + unary add.  8 waves/block handle 8 adjacent
// column tiles of one 16-row strip. Strip (2ch x 16 x 128 + halo) staged in
// LDS (pitch 140 -> conflict-free transposed reads). Computed transposed:
//   D_t(m=col, n=row) = Gband(16x28) x StripT(28x16), then D transposed back
// through LDS (pitch 17) for coalesced fused +unary stores.
// ---------------------------------------------------------------------------
#define HP 140
__global__ void __launch_bounds__(256)
crf_hblur_add(const float* __restrict__ tmp, const float* __restrict__ unary,
              float* __restrict__ out) {
  __shared__ float lin[2][16][HP];
  __shared__ float lout[8][2][16][17];

  const int tid  = threadIdx.x;
  const int lane = tid & 31;
  const int t    = tid >> 5;              // wave id == tile-in-group
  const int bid  = blockIdx.x;            // 16 * 32 * 4 blocks
  const int tcg  = bid & 3;
  const int tr   = (bid >> 2) & 31;
  const int b    = bid >> 7;
  const int cbase = tcg * 128 - 5;

  const float* t0 = tmp + (size_t)b * CHW;
  for (int i = tid; i < 2 * 16 * HP; i += 256) {
    int c  = i / (16 * HP);
    int rk = i - c * (16 * HP);
    int r  = rk / HP;
    int k  = rk - r * HP;
    int col = cbase + k;
    float v = 0.f;
    if ((unsigned)col < 512u) v = t0[c * HW + (tr * 16 + r) * Ww + col];
    lin[c][r][k] = v;                     // zero pad baked in (no NaN*0)
  }
  __syncthreads();

  const int n     = lane & 15;            // row within strip (B's N index)
  const int khalf = (lane < 16) ? 0 : 2;

  v2f A[7];
#pragma unroll
  for (int s = 0; s < 7; ++s) {
    int k0 = 4 * s + khalf;
    A[s].x = gband(n, k0);                // A row index m == lane&15 too
    A[s].y = gband(n, k0 + 1);
  }

  v8f acc0 = {0.f, 0.f, 0.f, 0.f, 0.f, 0.f, 0.f, 0.f};
  v8f acc1 = acc0;
#pragma unroll
  for (int s = 0; s < 7; ++s) {
    int kb = t * 16 + 4 * s + khalf;      // max 139 < HP
    v2f B0, B1;
    B0.x = lin[0][n][kb];
    B0.y = lin[0][n][kb + 1];
    B1.x = lin[1][n][kb];
    B1.y = lin[1][n][kb + 1];
    acc0 = wmma4(A[s], B0, acc0);
    acc1 = wmma4(A[s], B1, acc1);
  }

  // D_t: vgpr v -> col m = v + (lane<16?0:8), row n = lane&15
  const int mhalf = (lane < 16) ? 0 : 8;
#pragma unroll
  for (int v = 0; v < 8; ++v) {
    lout[t][0][n][v + mhalf] = acc0[v];
    lout[t][1][n][v + mhalf] = acc1[v];
  }
  __syncthreads();

  const float* u0 = unary + (size_t)b * CHW;
  float* o0 = out + (size_t)b * CHW;
#pragma unroll
  for (int i = 0; i < 16; ++i) {
    int idx = i * 32 + lane;              // 2*16*16 elems per wave
    int c   = idx >> 8;
    int r   = (idx >> 4) & 15;
    int mc  = idx & 15;
    size_t off = (size_t)c * HW + (size_t)(tr * 16 + r) * Ww +
                 (tcg * 128 + t * 16 + mc);
    o0[off] = lout[t][c][r][mc] + u0[off];
  }
}

// ---------------------------------------------------------------------------
extern "C" void kernel_launch(void* const* d_in, const int* in_sizes, int n_in,
                              void* d_out, int out_size, void* d_ws,
                              size_t ws_size, hipStream_t stream) {
  const float* x  = (const float*)d_in[0];   // (16,3,512,512)
  const float* w  = (const float*)d_in[1];   // (2,3,3,3)
  const float* cb = (const float*)d_in[2];   // (2,)
  float* cur   = (float*)d_out;              // (16,2,512,512)
  float* unary = (float*)d_ws;               // 33.5 MB
  float* tmp   = unary + (size_t)16 * CHW;   // 33.5 MB

  conv3x3_kernel<<<16384, 256, 0, stream>>>(x, w, cb, unary);

  for (int it = 0; it < 5; ++it) {
    const float* src = (it == 0) ? unary : cur;
    crf_vblur_softmax<<<2048, 256, 0, stream>>>(src, tmp);   // 16384 tiles
    crf_hblur_add<<<2048, 256, 0, stream>>>(tmp, unary, cur);
  }
}